// Model_48619029791057
// MI455X (gfx1250) — compile-verified
//
#include <hip/hip_runtime.h>
#include <math.h>

#define LL 128
#define BB 32
#define EE 512
#define HH 512
#define H2 1024
#define VV 32000

typedef __attribute__((ext_vector_type(2))) float v2f;
typedef __attribute__((ext_vector_type(8))) float v8f;

__device__ inline v8f zero8() {
    v8f z;
    #pragma unroll
    for (int i = 0; i < 8; i++) z[i] = 0.f;
    return z;
}
__device__ inline float sigm(float x) { return 1.f / (1.f + expf(-x)); }

// ---------------------------------------------------------------------------
// Wave-level fp32 WMMA: computes two 16x16 tiles (rows 0-15 from A0, rows
// 16-31 from A1) of  C += A(32xK) * W^T  where A is row-major (lda) and W is
// row-major N x K (ldw).  B fragment is shared between the two M-tiles.
// Layout per CDNA5 ISA 7.12.2 (32-bit A 16x4 / B 4x16):
//   lane = m + 16*(k_local/2), vgpr = k_local%2  -> per-lane float2 loads.
// ---------------------------------------------------------------------------
__device__ inline void wmma_2tiles(const float* A0, const float* A1, int lda,
                                   const float* W, int ldw, int K,
                                   v8f& acc0, v8f& acc1) {
    int lane = threadIdx.x & 31;
    int r  = lane & 15;          // m for A, n for B
    int kh = (lane >> 4) << 1;   // 0 or 2 (k-half within the 4-wide step)
    const float* a0p = A0 + (size_t)r * lda + kh;
    const float* a1p = A1 + (size_t)r * lda + kh;
    const float* bp  = W  + (size_t)r * ldw + kh;
    for (int k = 0; k < K; k += 4) {
        v2f b = *(const v2f*)(bp + k);
        v2f a = *(const v2f*)(a0p + k);
        acc0 = __builtin_amdgcn_wmma_f32_16x16x4_f32(false, a, false, b,
                                                     (short)0, acc0, false, false);
        a = *(const v2f*)(a1p + k);
        acc1 = __builtin_amdgcn_wmma_f32_16x16x4_f32(false, a, false, b,
                                                     (short)0, acc1, false, false);
    }
}

// Store both M-tiles of a 32x16 column strip.  C points at (row0, col0).
__device__ inline void store2(float* C, int ldc, v8f a0, v8f a1) {
    int lane = threadIdx.x & 31;
    int n  = lane & 15;
    int mb = (lane >> 4) * 8;
    #pragma unroll
    for (int i = 0; i < 8; i++) C[(size_t)(mb + i) * ldc + n] = a0[i];
    #pragma unroll
    for (int i = 0; i < 8; i++) C[(size_t)(16 + mb + i) * ldc + n] = a1[i];
}

// ---------------------------------------------------------------------------
// init: zero h0/c, loss = 0, inp = SOS(65)
// ---------------------------------------------------------------------------
__global__ void init_kernel(float* h0, float* cbuf, float* loss, int* inp) {
    int idx = blockIdx.x * blockDim.x + threadIdx.x;
    if (idx < BB * H2) h0[idx] = 0.f;
    int idx2 = idx - BB * H2;
    if (idx2 >= 0 && idx2 < BB * H2) cbuf[idx2] = 0.f;
    if (idx == 0) *loss = 0.f;
    if (idx < BB) inp[idx] = 65;
}

// ---------------------------------------------------------------------------
// Encoder: only batch row 0 matters (reference broadcasts out[0]).  Single
// persistent workgroup, two LSTM cells (f/b) per step, 128 steps.
// Writes encoder outputs TRANSPOSED: encT[(d)*128 + t], d in [0,1024).
// ---------------------------------------------------------------------------
__global__ void encoder_kernel(const int* __restrict__ x, const float* __restrict__ emb_enc,
    const float* __restrict__ Wih_f, const float* __restrict__ Whh_f,
    const float* __restrict__ bih_f, const float* __restrict__ bhh_f,
    const float* __restrict__ Wih_b, const float* __restrict__ Whh_b,
    const float* __restrict__ bih_b, const float* __restrict__ bhh_b,
    float* __restrict__ encT) {
    __shared__ float e[EE], hf[HH], cf[HH], hb[HH], cb[HH];
    __shared__ float gf[4 * HH], gb[4 * HH];
    int tid = threadIdx.x;  // 1024 threads
    if (tid < HH) { hf[tid] = 0.f; cf[tid] = 0.f; hb[tid] = 0.f; cb[tid] = 0.f; }
    __syncthreads();
    for (int t = 0; t < LL; t++) {
        int id = x[t];  // x[0][t]
        if (tid < EE) e[tid] = emb_enc[(size_t)id * EE + tid];
        __syncthreads();
        for (int half = 0; half < 2; half++) {
            int g = tid + half * 1024;  // gate row in [0,2048)
            const float* wif = Wih_f + (size_t)g * EE;
            const float* whf = Whh_f + (size_t)g * HH;
            const float* wib = Wih_b + (size_t)g * EE;
            const float* whb = Whh_b + (size_t)g * HH;
            float sf = bih_f[g] + bhh_f[g];
            float sb = bih_b[g] + bhh_b[g];
            #pragma unroll 4
            for (int k = 0; k < EE; k++) {
                float ek = e[k];
                sf += wif[k] * ek + whf[k] * hf[k];
                sb += wib[k] * ek + whb[k] * hb[k];
            }
            gf[g] = sf; gb[g] = sb;
        }
        __syncthreads();
        if (tid < HH) {
            int j = tid;
            float i1 = gf[j], f1 = gf[HH + j], g1 = gf[2 * HH + j], o1 = gf[3 * HH + j];
            float cn = sigm(f1) * cf[j] + sigm(i1) * tanhf(g1);
            float hn = sigm(o1) * tanhf(cn);
            cf[j] = cn; hf[j] = hn;
            encT[(size_t)j * LL + t] = hn;
            float i2 = gb[j], f2 = gb[HH + j], g2 = gb[2 * HH + j], o2 = gb[3 * HH + j];
            float cn2 = sigm(f2) * cb[j] + sigm(i2) * tanhf(g2);
            float hn2 = sigm(o2) * tanhf(cn2);
            cb[j] = cn2; hb[j] = hn2;
            encT[(size_t)(HH + j) * LL + t] = hn2;
        }
        __syncthreads();
    }
}

// ---------------------------------------------------------------------------
// Decoder step kernels
// ---------------------------------------------------------------------------
__global__ void gather_kernel(const int* __restrict__ inp, const float* __restrict__ emb_dec,
                              float* __restrict__ ebuf) {
    int b = blockIdx.x;
    const float* row = emb_dec + (size_t)inp[b] * H2;
    for (int j = threadIdx.x; j < H2; j += blockDim.x) ebuf[b * H2 + j] = row[j];
}

// attention logits (WMMA) -> softmax -> context (WMMA against encT) ; 1 WG
__global__ void attn_ctx_kernel(const float* __restrict__ ebuf, const float* __restrict__ hin,
                                const float* __restrict__ W_attn, const float* __restrict__ b_attn,
                                const float* __restrict__ encT, float* __restrict__ ctx) {
    __shared__ float aw[BB][LL];
    int tid = threadIdx.x;
    int w = tid >> 5, lane = tid & 31;
    int n = lane & 15, mb = (lane >> 4) * 8;
    {
        int col0 = w * 16;  // 8 waves * 16 = 128 = L
        v8f a0 = zero8(), a1 = zero8();
        // attn_in = [e, h]: split-K accumulate against the two halves of W_attn
        wmma_2tiles(ebuf, ebuf + 16 * H2, H2, W_attn + (size_t)col0 * (2 * H2), 2 * H2, H2, a0, a1);
        wmma_2tiles(hin,  hin  + 16 * H2, H2, W_attn + (size_t)col0 * (2 * H2) + H2, 2 * H2, H2, a0, a1);
        float bias = b_attn[col0 + n];
        #pragma unroll
        for (int i = 0; i < 8; i++) {
            aw[mb + i][col0 + n]      = a0[i] + bias;
            aw[16 + mb + i][col0 + n] = a1[i] + bias;
        }
    }
    __syncthreads();
    if (tid < BB) {  // row-wise softmax over 128
        float mx = -INFINITY;
        for (int j = 0; j < LL; j++) mx = fmaxf(mx, aw[tid][j]);
        float s = 0.f;
        for (int j = 0; j < LL; j++) { float v = expf(aw[tid][j] - mx); aw[tid][j] = v; s += v; }
        float inv = 1.f / s;
        for (int j = 0; j < LL; j++) aw[tid][j] *= inv;
    }
    __syncthreads();
    const float* A0 = &aw[0][0];
    const float* A1 = &aw[16][0];
    for (int nt = w * 8; nt < w * 8 + 8; nt++) {  // 64 column tiles over H2=1024
        int col0 = nt * 16;
        v8f a0 = zero8(), a1 = zero8();
        wmma_2tiles(A0, A1, LL, encT + (size_t)col0 * LL, LL, LL, a0, a1);
        store2(ctx + col0, H2, a0, a1);
    }
}

// comb = relu([e, ctx] @ W_comb^T + b_comb) ; 8 WGs x 128 cols
__global__ void comb_kernel(const float* __restrict__ ebuf, const float* __restrict__ ctx,
                            const float* __restrict__ W_comb, const float* __restrict__ b_comb,
                            float* __restrict__ comb) {
    int tid = threadIdx.x;
    int w = tid >> 5, lane = tid & 31, n = lane & 15;
    int col0 = blockIdx.x * 128 + w * 16;
    v8f a0 = zero8(), a1 = zero8();
    wmma_2tiles(ebuf, ebuf + 16 * H2, H2, W_comb + (size_t)col0 * (2 * H2), 2 * H2, H2, a0, a1);
    wmma_2tiles(ctx,  ctx  + 16 * H2, H2, W_comb + (size_t)col0 * (2 * H2) + H2, 2 * H2, H2, a0, a1);
    float bias = b_comb[col0 + n];
    #pragma unroll
    for (int i = 0; i < 8; i++) { a0[i] = fmaxf(a0[i] + bias, 0.f); a1[i] = fmaxf(a1[i] + bias, 0.f); }
    store2(comb + col0, H2, a0, a1);
}

// gates = comb@Wih_d^T + h@Whh_d^T + biases (WMMA), then fused LSTM pointwise.
// Each WG owns a 64-wide h-slice -> all 4 gates for those columns are local.
__global__ void lstm_kernel(const float* __restrict__ comb, const float* __restrict__ hin,
                            const float* __restrict__ Wih_d, const float* __restrict__ Whh_d,
                            const float* __restrict__ bih_d, const float* __restrict__ bhh_d,
                            float* __restrict__ cbuf, float* __restrict__ hout) {
    __shared__ float gbuf[BB][256];  // [row][gate*64 + j]
    int g = blockIdx.x;              // 16 WGs, h-slice [g*64, g*64+64)
    int tid = threadIdx.x;
    int w = tid >> 5, lane = tid & 31, n = lane & 15, mb = (lane >> 4) * 8;
    for (int ct = 2 * w; ct < 2 * w + 2; ct++) {  // 16 column tiles / 8 waves
        int gi = ct >> 2, tt = ct & 3;
        int col0 = gi * H2 + g * 64 + tt * 16;    // gate column in [0,4096)
        v8f a0 = zero8(), a1 = zero8();
        wmma_2tiles(comb, comb + 16 * H2, H2, Wih_d + (size_t)col0 * H2, H2, H2, a0, a1);
        wmma_2tiles(hin,  hin  + 16 * H2, H2, Whh_d + (size_t)col0 * H2, H2, H2, a0, a1);
        float bias = bih_d[col0 + n] + bhh_d[col0 + n];
        int lc = gi * 64 + tt * 16 + n;
        #pragma unroll
        for (int i = 0; i < 8; i++) {
            gbuf[mb + i][lc]      = a0[i] + bias;
            gbuf[16 + mb + i][lc] = a1[i] + bias;
        }
    }
    __syncthreads();
    for (int r = tid; r < BB * 64; r += blockDim.x) {
        int m = r >> 6, j = r & 63;
        float ig = gbuf[m][j], fg = gbuf[m][64 + j], gg = gbuf[m][128 + j], og = gbuf[m][192 + j];
        int J = g * 64 + j;
        float cp = cbuf[m * H2 + J];
        float cn = sigm(fg) * cp + sigm(ig) * tanhf(gg);
        cbuf[m * H2 + J] = cn;
        hout[m * H2 + J] = sigm(og) * tanhf(cn);
    }
}

// logits = h @ W_out^T + b_out ; 250 WGs x 128 cols, WMMA, W_out L2-resident
__global__ void logits_kernel(const float* __restrict__ h, const float* __restrict__ W_out,
                              const float* __restrict__ b_out, float* __restrict__ logits) {
    int tid = threadIdx.x;
    int w = tid >> 5, lane = tid & 31, n = lane & 15;
    int col0 = blockIdx.x * 128 + w * 16;
    v8f a0 = zero8(), a1 = zero8();
    wmma_2tiles(h, h + 16 * H2, H2, W_out + (size_t)col0 * H2, H2, H2, a0, a1);
    float bias = b_out[col0 + n];
    #pragma unroll
    for (int i = 0; i < 8; i++) { a0[i] += bias; a1[i] += bias; }
    store2(logits + col0, VV, a0, a1);
}

// per-row: max + argmax (lowest-index tie-break), logsumexp, NLL, greedy token
__global__ void softmax_kernel(const float* __restrict__ logits, const int* __restrict__ tgt,
                               int t, float* __restrict__ nll_rows, int* __restrict__ inp) {
    __shared__ float sv[256];
    __shared__ int   si[256];
    __shared__ float ss[256];
    int b = blockIdx.x, tid = threadIdx.x;
    const float* row = logits + (size_t)b * VV;
    float mx = -INFINITY; int mi = 0;
    for (int j = tid; j < VV; j += 256) {
        float v = row[j];
        if (v > mx || (v == mx && j < mi)) { mx = v; mi = j; }
    }
    sv[tid] = mx; si[tid] = mi;
    __syncthreads();
    for (int s = 128; s > 0; s >>= 1) {
        if (tid < s) {
            float v2 = sv[tid + s]; int i2 = si[tid + s];
            if (v2 > sv[tid] || (v2 == sv[tid] && i2 < si[tid])) { sv[tid] = v2; si[tid] = i2; }
        }
        __syncthreads();
    }
    float rmax = sv[0]; int amax = si[0];
    float sum = 0.f;
    for (int j = tid; j < VV; j += 256) sum += expf(row[j] - rmax);
    ss[tid] = sum;
    __syncthreads();
    for (int s = 128; s > 0; s >>= 1) {
        if (tid < s) ss[tid] += ss[tid + s];
        __syncthreads();
    }
    if (tid == 0) {
        int tg = tgt[b * LL + t];
        float lp = row[tg] - rmax - logf(ss[0]);
        nll_rows[b] = -lp / (float)BB;  // mean over batch
        inp[b] = amax;                  // greedy feedback for step t+1
    }
}

__global__ void loss_kernel(const float* __restrict__ nll_rows, float* __restrict__ loss) {
    if (threadIdx.x == 0 && blockIdx.x == 0) {
        float s = 0.f;
        for (int b = 0; b < BB; b++) s += nll_rows[b];  // fixed order: deterministic
        *loss += s;
    }
}

__global__ void final_kernel(const float* __restrict__ loss, float* __restrict__ out) {
    if (threadIdx.x == 0 && blockIdx.x == 0) {
        out[0] = *loss;
        out[1] = *loss / (float)LL;
    }
}

// ---------------------------------------------------------------------------
extern "C" void kernel_launch(void* const* d_in, const int* in_sizes, int n_in,
                              void* d_out, int out_size, void* d_ws, size_t ws_size,
                              hipStream_t stream) {
    (void)in_sizes; (void)n_in; (void)out_size; (void)ws_size;
    const int*   x       = (const int*)d_in[0];
    const int*   tgt     = (const int*)d_in[1];
    const float* emb_enc = (const float*)d_in[4];
    const float* Wih_f   = (const float*)d_in[5];
    const float* Whh_f   = (const float*)d_in[6];
    const float* bih_f   = (const float*)d_in[7];
    const float* bhh_f   = (const float*)d_in[8];
    const float* Wih_b   = (const float*)d_in[9];
    const float* Whh_b   = (const float*)d_in[10];
    const float* bih_b   = (const float*)d_in[11];
    const float* bhh_b   = (const float*)d_in[12];
    const float* emb_dec = (const float*)d_in[13];
    const float* W_attn  = (const float*)d_in[14];
    const float* b_attn  = (const float*)d_in[15];
    const float* W_comb  = (const float*)d_in[16];
    const float* b_comb  = (const float*)d_in[17];
    const float* Wih_d   = (const float*)d_in[18];
    const float* Whh_d   = (const float*)d_in[19];
    const float* bih_d   = (const float*)d_in[20];
    const float* bhh_d   = (const float*)d_in[21];
    const float* W_out   = (const float*)d_in[22];
    const float* b_out   = (const float*)d_in[23];

    float* ws   = (float*)d_ws;
    float* ENCT = ws;                   // 1024 x 128 (transposed encoder outputs)
    float* H0   = ENCT + 131072;        // 32 x 1024
    float* H1   = H0 + 32768;           // 32 x 1024 (ping-pong)
    float* CC   = H1 + 32768;           // 32 x 1024
    float* EB   = CC + 32768;           // 32 x 1024 embedded input
    float* CTX  = EB + 32768;           // 32 x 1024
    float* COMB = CTX + 32768;          // 32 x 1024
    float* LOG  = COMB + 32768;         // 32 x 32000
    float* NLL  = LOG + 1024000;        // 32
    float* LOSS = NLL + 32;             // 1
    int*   INP  = (int*)(LOSS + 1);     // 32

    init_kernel<<<(2 * BB * H2 + 255) / 256, 256, 0, stream>>>(H0, CC, LOSS, INP);
    encoder_kernel<<<1, 1024, 0, stream>>>(x, emb_enc, Wih_f, Whh_f, bih_f, bhh_f,
                                           Wih_b, Whh_b, bih_b, bhh_b, ENCT);
    for (int t = 0; t < LL; t++) {
        float* hin  = (t & 1) ? H1 : H0;
        float* hout = (t & 1) ? H0 : H1;
        gather_kernel  <<<BB, 256, 0, stream>>>(INP, emb_dec, EB);
        attn_ctx_kernel<<<1, 256, 0, stream>>>(EB, hin, W_attn, b_attn, ENCT, CTX);
        comb_kernel    <<<8, 256, 0, stream>>>(EB, CTX, W_comb, b_comb, COMB);
        lstm_kernel    <<<16, 256, 0, stream>>>(COMB, hin, Wih_d, Whh_d, bih_d, bhh_d, CC, hout);
        logits_kernel  <<<250, 256, 0, stream>>>(hout, W_out, b_out, LOG);
        softmax_kernel <<<BB, 256, 0, stream>>>(LOG, tgt, t, NLL, INP);
        loss_kernel    <<<1, 1, 0, stream>>>(NLL, LOSS);
    }
    final_kernel<<<1, 1, 0, stream>>>(LOSS, (float*)d_out);
}